// GCN_LSTM_67224828117588
// MI455X (gfx1250) — compile-verified
//
#include <hip/hip_runtime.h>

typedef __attribute__((ext_vector_type(2))) float v2f;
typedef __attribute__((ext_vector_type(8))) float v8f;

#define WAVES_PER_BLOCK 4
#define TILE 16
#define HSTR 132   // LDS row stride (dwords) for H scratch: conflict-free A loads
#define H1STR 68   // LDS row stride for layer-1 activations

#define LOG2E 1.44269504088896340736f

// Branchless sigmoid: rcp(1 + exp2(-x*log2e)).
// x->+inf: exp2->0, rcp(1)=1.  x->-inf: exp2->inf, rcp(inf)=0.
__device__ __forceinline__ float fast_sigmoid(float v) {
    float e = __builtin_amdgcn_exp2f(-v * LOG2E);
    return __builtin_amdgcn_rcpf(1.0f + e);
}

// Native v_tanh_f32 when the toolchain exposes it, else branchless identity
// tanh(x) = 1 - 2*rcp(1 + exp2(2x*log2e)).
__device__ __forceinline__ float fast_tanh(float v) {
#if __has_builtin(__builtin_amdgcn_tanhf)
    return __builtin_amdgcn_tanhf(v);
#else
    float e = __builtin_amdgcn_exp2f(2.0f * v * LOG2E);
    return 1.0f - 2.0f * __builtin_amdgcn_rcpf(1.0f + e);
#endif
}

__device__ __forceinline__ v8f wmma4_f32(v2f a, v2f b, v8f c) {
    // D = A(16x4 f32) x B(4x16 f32) + C(16x16 f32)
    return __builtin_amdgcn_wmma_f32_16x16x4_f32(false, a, false, b, (short)0, c,
                                                 false, false);
}

__global__ __launch_bounds__(WAVES_PER_BLOCK * 32)
void gclstm_mlp_kernel(const float* __restrict__ x,
                       const float* __restrict__ Wi, const float* __restrict__ Wc,
                       const float* __restrict__ Wo,
                       const float* __restrict__ bci, const float* __restrict__ bcc,
                       const float* __restrict__ bco,
                       const float* __restrict__ bgi, const float* __restrict__ bgc,
                       const float* __restrict__ bgo,
                       const float* __restrict__ wco,
                       const float* __restrict__ M1, const float* __restrict__ bm1,
                       const float* __restrict__ M2, const float* __restrict__ bm2,
                       const float* __restrict__ M3, const float* __restrict__ bm3,
                       float* __restrict__ out, int n_rows, int n_tiles)
{
    __shared__ float lds[WAVES_PER_BLOCK * (TILE * HSTR + TILE * H1STR)];

    const int tid   = threadIdx.x;
    const int wave  = tid >> 5;
    const int lane  = tid & 31;
    const int colh  = lane & 15;               // column / row-in-tile index
    const int half2 = (lane < 16) ? 0 : 2;     // K offset for upper half-wave (A/B layout)
    const int half8 = (lane < 16) ? 0 : 8;     // M offset for upper half-wave (C/D layout)

    float* Hs  = lds + wave * (TILE * HSTR + TILE * H1STR);
    float* h1s = Hs + TILE * HSTR;

    // ---- per-wave invariant per-lane parameters (hidden col = 16n + colh) ----
    float biasI[8], biasC[8], biasO[8], wcov[8];
#pragma unroll
    for (int n = 0; n < 8; ++n) {
        int c = 16 * n + colh;
        biasI[n] = bci[c] + bgi[c];
        biasC[n] = bcc[c] + bgc[c];
        biasO[n] = bco[c] + bgo[c];
        wcov[n]  = wco[c];
    }
    float bm1v[4];
#pragma unroll
    for (int n = 0; n < 4; ++n) bm1v[n] = bm1[16 * n + colh];
    float bm2v[2], m3v[2];
#pragma unroll
    for (int n = 0; n < 2; ++n) {
        bm2v[n] = bm2[16 * n + colh];
        m3v[n]  = M3[16 * n + colh];
    }
    const float bm3s = bm3[0];

    const int waveGlobal = blockIdx.x * WAVES_PER_BLOCK + wave;
    const int waveStride = gridDim.x * WAVES_PER_BLOCK;

    for (int tile = waveGlobal; tile < n_tiles; tile += waveStride) {
        const int rowBase = tile * TILE;
        int myRow = rowBase + colh;
        if (myRow > n_rows - 1) myRow = n_rows - 1;   // duplicate last row in tail tile

        // ---- A operand: x tile 16x16 in f32 A-layout (4 K-steps of 4) ----
        v2f ax[4];
        const float* xrow = x + (size_t)myRow * 16 + half2;
#pragma unroll
        for (int t = 0; t < 4; ++t)
            ax[t] = *(const v2f*)(xrow + 4 * t);

        // ================= phase 1: LSTM gates, 8 hidden tiles =================
#pragma unroll
        for (int n = 0; n < 8; ++n) {
            const int col = 16 * n + colh;
            const float* pWi = Wi + half2 * 128 + col;
            const float* pWc = Wc + half2 * 128 + col;
            const float* pWo = Wo + half2 * 128 + col;
            v8f ai = {}, at = {}, ao = {};
#pragma unroll
            for (int t = 0; t < 4; ++t) {
                v2f bI, bC, bO;
                bI.x = pWi[(4 * t) * 128]; bI.y = pWi[(4 * t + 1) * 128];
                bC.x = pWc[(4 * t) * 128]; bC.y = pWc[(4 * t + 1) * 128];
                bO.x = pWo[(4 * t) * 128]; bO.y = pWo[(4 * t + 1) * 128];
                ai = wmma4_f32(ax[t], bI, ai);
                at = wmma4_f32(ax[t], bC, at);
                ao = wmma4_f32(ax[t], bO, ao);
            }
            // element-wise LSTM in C/D layout; store relu(H) to LDS (row-major)
#pragma unroll
            for (int r = 0; r < 8; ++r) {
                float I  = fast_sigmoid(ai[r] + biasI[n]);
                float T  = fast_tanh(at[r] + biasC[n]);
                float Cv = I * T;                               // C_old == 0
                float O  = fast_sigmoid(ao[r] + biasO[n] + wcov[n] * Cv);
                float Hv = O * fast_tanh(Cv);
                Hs[(r + half8) * HSTR + col] = fmaxf(Hv, 0.0f); // relu(H)
            }
        }

        // ================= phase 2: MLP layer 1 (128 -> 64) =================
#pragma unroll
        for (int n = 0; n < 4; ++n) {
            const int col = 16 * n + colh;
            const float* pM1 = M1 + half2 * 64 + col;
            v8f acc = {};
#pragma unroll 8
            for (int t = 0; t < 32; ++t) {      // K = 128
                v2f a = *(const v2f*)(Hs + colh * HSTR + 4 * t + half2);
                v2f b;
                b.x = pM1[(4 * t) * 64]; b.y = pM1[(4 * t + 1) * 64];
                acc = wmma4_f32(a, b, acc);
            }
#pragma unroll
            for (int r = 0; r < 8; ++r)
                h1s[(r + half8) * H1STR + col] = fmaxf(acc[r] + bm1v[n], 0.0f);
        }

        // ========= phase 3: MLP layer 2 (64 -> 32) + layer 3 (32 -> 1) =========
        float red[8];
#pragma unroll
        for (int r = 0; r < 8; ++r) red[r] = 0.0f;
#pragma unroll
        for (int n = 0; n < 2; ++n) {
            const int col = 16 * n + colh;
            const float* pM2 = M2 + half2 * 32 + col;
            v8f acc = {};
#pragma unroll
            for (int t = 0; t < 16; ++t) {      // K = 64
                v2f a = *(const v2f*)(h1s + colh * H1STR + 4 * t + half2);
                v2f b;
                b.x = pM2[(4 * t) * 32]; b.y = pM2[(4 * t + 1) * 32];
                acc = wmma4_f32(a, b, acc);
            }
#pragma unroll
            for (int r = 0; r < 8; ++r)
                red[r] += fmaxf(acc[r] + bm2v[n], 0.0f) * m3v[n];
        }

        // reduce over 16 hidden columns inside each half-wave (C/D layout:
        // lanes 0-15 hold node r, lanes 16-31 hold node r+8)
#pragma unroll
        for (int r = 0; r < 8; ++r) {
#pragma unroll
            for (int s = 1; s < 16; s <<= 1)
                red[r] += __shfl_xor(red[r], s, 32);
        }
        if (colh == 0) {
#pragma unroll
            for (int r = 0; r < 8; ++r) {
                int orow = rowBase + r + half8;
                if (orow < n_rows) out[orow] = red[r] + bm3s;
            }
        }
    }
}

extern "C" void kernel_launch(void* const* d_in, const int* in_sizes, int n_in,
                              void* d_out, int out_size, void* d_ws, size_t ws_size,
                              hipStream_t stream) {
    (void)n_in; (void)d_ws; (void)ws_size; (void)out_size;
    // setup_inputs() dict order:
    //  0 x, 1 edge_index, 2 edge_weight, 3 W_i, 4 W_f, 5 W_c, 6 W_o,
    //  7..10 Uc_*, 11 bc_i, 12 bc_f, 13 bc_c, 14 bc_o,
    //  15 w_ci, 16 w_cf, 17 w_co, 18 b_i, 19 b_f, 20 b_c, 21 b_o,
    //  22 M1, 23 bm1, 24 M2, 25 bm2, 26 M3, 27 bm3
    const float* x   = (const float*)d_in[0];
    const float* Wi  = (const float*)d_in[3];
    const float* Wc  = (const float*)d_in[5];
    const float* Wo  = (const float*)d_in[6];
    const float* bci = (const float*)d_in[11];
    const float* bcc = (const float*)d_in[13];
    const float* bco = (const float*)d_in[14];
    const float* wco = (const float*)d_in[17];
    const float* bgi = (const float*)d_in[18];
    const float* bgc = (const float*)d_in[20];
    const float* bgo = (const float*)d_in[21];
    const float* M1  = (const float*)d_in[22];
    const float* bm1 = (const float*)d_in[23];
    const float* M2  = (const float*)d_in[24];
    const float* bm2 = (const float*)d_in[25];
    const float* M3  = (const float*)d_in[26];
    const float* bm3 = (const float*)d_in[27];
    float* out = (float*)d_out;

    const int n_rows  = in_sizes[0] / 16;           // N nodes (x is [N,16])
    const int n_tiles = (n_rows + TILE - 1) / TILE; // 31250 for N=500000

    int blocks = (n_tiles + WAVES_PER_BLOCK - 1) / WAVES_PER_BLOCK;
    if (blocks > 2048) blocks = 2048;               // grid-stride over tiles

    gclstm_mlp_kernel<<<blocks, WAVES_PER_BLOCK * 32, 0, stream>>>(
        x, Wi, Wc, Wo, bci, bcc, bco, bgi, bgc, bgo, wco,
        M1, bm1, M2, bm2, M3, bm3, out, n_rows, n_tiles);
}